// computeAttentionWeightProj_48782238548114
// MI455X (gfx1250) — compile-verified
//
#include <hip/hip_runtime.h>

#define HEADS  8
#define BATCH  4
#define SEQ    2048
#define CDIM   32
#define WINDOW 10

#define WAVES  8              // waves per block
#define STILES (SEQ / 16)     // 128 query tiles per (h,n)

typedef __attribute__((ext_vector_type(2))) float v2f;
typedef __attribute__((ext_vector_type(8))) float v8f;

__device__ __forceinline__ v8f wmma4(v2f a, v2f b, v8f c) {
  // V_WMMA_F32_16X16X4_F32 : D = A(16x4,f32) * B(4x16,f32) + C(16x16,f32)
  return __builtin_amdgcn_wmma_f32_16x16x4_f32(
      /*neg_a=*/false, a, /*neg_b=*/false, b,
      /*c_mod=*/(short)0, c, /*reuse_a=*/false, /*reuse_b=*/false);
}

__global__ __launch_bounds__(32 * WAVES)
void band_attn_wmma_kernel(const float* __restrict__ Q,
                           const float* __restrict__ K,
                           const float* __restrict__ V,
                           const float* __restrict__ G,
                           const unsigned char* __restrict__ seqMask,
                           float* __restrict__ out)
{
  // per-wave probability staging buffer (C/D layout -> A layout transpose)
  __shared__ float Abuf[WAVES][16][49];   // stride 49: coprime with 64 banks

  const int wave  = (int)(threadIdx.x >> 5);
  const int lane  = (int)(threadIdx.x & 31);
  const int lhalf = lane & 15;
  const int hi    = lane >> 4;            // 0: lanes 0-15, 1: lanes 16-31

  const int tileId = (int)blockIdx.x * WAVES + wave;
  const int st = tileId & (STILES - 1);
  const int hn = tileId / STILES;
  const int n  = hn & (BATCH - 1);
  const int h  = hn / BATCH;
  const int s0 = st * 16;

  const size_t hnBase = (size_t)(h * BATCH + n) * SEQ * CDIM;
  const float  scale  = 0.17677669529663687f; // 1/sqrt(CDIM)

  // ---------------- scores: S = Q K^T over 3 aligned key tiles ----------------
  v8f Sacc[3];
  Sacc[0] = (v8f){}; Sacc[1] = (v8f){}; Sacc[2] = (v8f){};

  const int t_of0 = s0 - 16 + lhalf;
  const int t_of1 = s0      + lhalf;
  const int t_of2 = s0 + 16 + lhalf;

  {
    const float* qrow = Q + hnBase + (size_t)(s0 + lhalf) * CDIM + 2 * hi;
    const int tm = (t_of0 < 0) ? 0 : t_of0;
    const int tp = (t_of2 >= SEQ) ? (SEQ - 1) : t_of2;
    const float* krow0 = K + hnBase + (size_t)tm    * CDIM + 2 * hi;
    const float* krow1 = K + hnBase + (size_t)t_of1 * CDIM + 2 * hi;
    const float* krow2 = K + hnBase + (size_t)tp    * CDIM + 2 * hi;
#pragma unroll
    for (int c4 = 0; c4 < CDIM; c4 += 4) {
      v2f a  = *(const v2f*)(qrow  + c4);
      v2f b0 = *(const v2f*)(krow0 + c4);
      v2f b1 = *(const v2f*)(krow1 + c4);
      v2f b2 = *(const v2f*)(krow2 + c4);
      Sacc[0] = wmma4(a, b0, Sacc[0]);
      Sacc[1] = wmma4(a, b1, Sacc[1]);
      Sacc[2] = wmma4(a, b2, Sacc[2]);
    }
  }

  // per-lane column validity (this lane's column within each key tile)
  bool cmask[3];
  {
    const int tarr[3] = { t_of0, t_of1, t_of2 };
#pragma unroll
    for (int kt = 0; kt < 3; ++kt) {
      int t = tarr[kt];
      bool v = (t >= 0) && (t < SEQ);
      if (v) v = (seqMask[(size_t)n * SEQ + t] == 0);
      cmask[kt] = v;
    }
  }

  // ------------- band mask + softmax per row, write probs to LDS -------------
  const int tarr[3] = { t_of0, t_of1, t_of2 };
#pragma unroll
  for (int r = 0; r < 8; ++r) {
    const int srow = s0 + r + 8 * hi;      // C/D layout: VGPR r holds rows r, r+8
    float sc[3];
#pragma unroll
    for (int kt = 0; kt < 3; ++kt) {
      int d = srow - tarr[kt]; d = (d < 0) ? -d : d;
      const bool valid = cmask[kt] && (d <= WINDOW);
      sc[kt] = valid ? (Sacc[kt][r] * scale) : -1e30f;
    }
    float m = fmaxf(sc[0], fmaxf(sc[1], sc[2]));
#pragma unroll
    for (int off = 8; off; off >>= 1) m = fmaxf(m, __shfl_xor(m, off, 32));
    float p0 = __expf(sc[0] - m);
    float p1 = __expf(sc[1] - m);
    float p2 = __expf(sc[2] - m);
    float sum = p0 + p1 + p2;
#pragma unroll
    for (int off = 8; off; off >>= 1) sum += __shfl_xor(sum, off, 32);
    const float inv = 1.0f / sum;
    Abuf[wave][r + 8 * hi][ 0 + lhalf] = p0 * inv;
    Abuf[wave][r + 8 * hi][16 + lhalf] = p1 * inv;
    Abuf[wave][r + 8 * hi][32 + lhalf] = p2 * inv;
  }

  // ------------------------ O = A(16x48) * V(48x32) --------------------------
  v8f O0 = (v8f){}, O1 = (v8f){};
  const float* vbase = V + hnBase;
#pragma unroll
  for (int kk = 0; kk < 12; ++kk) {
    v2f a;
    a.x = Abuf[wave][lhalf][kk * 4 + 2 * hi + 0];
    a.y = Abuf[wave][lhalf][kk * 4 + 2 * hi + 1];
    int t0 = s0 - 16 + kk * 4 + 2 * hi;     // V rows for this lane's B pair
    int r0 = t0;     if (r0 < 0) r0 = 0; else if (r0 > SEQ - 1) r0 = SEQ - 1;
    int r1 = t0 + 1; if (r1 < 0) r1 = 0; else if (r1 > SEQ - 1) r1 = SEQ - 1;
    const float* vr0 = vbase + (size_t)r0 * CDIM + lhalf;
    const float* vr1 = vbase + (size_t)r1 * CDIM + lhalf;
    v2f b0; b0.x = vr0[0];  b0.y = vr1[0];   // channels 0..15
    v2f b1; b1.x = vr0[16]; b1.y = vr1[16];  // channels 16..31
    O0 = wmma4(a, b0, O0);
    O1 = wmma4(a, b1, O1);
  }

  // --------------------- gate with G and store [n,s,h,c] ---------------------
#pragma unroll
  for (int r = 0; r < 8; ++r) {
    const int srow = s0 + r + 8 * hi;
    const size_t gi = hnBase + (size_t)srow * CDIM + lhalf;
    const size_t oi = (((size_t)n * SEQ + srow) * HEADS + h) * CDIM + lhalf;
    out[oi]      = O0[r] * G[gi];
    out[oi + 16] = O1[r] * G[gi + 16];
  }
}

extern "C" void kernel_launch(void* const* d_in, const int* in_sizes, int n_in,
                              void* d_out, int out_size, void* d_ws, size_t ws_size,
                              hipStream_t stream) {
  (void)in_sizes; (void)n_in; (void)out_size; (void)d_ws; (void)ws_size;
  const float*         Q    = (const float*)d_in[0];
  const float*         K    = (const float*)d_in[1];
  const float*         V    = (const float*)d_in[2];
  const float*         G    = (const float*)d_in[3];
  const unsigned char* mask = (const unsigned char*)d_in[4];
  float* out = (float*)d_out;

  const int tiles  = HEADS * BATCH * STILES;   // 4096 wave-tiles
  const int blocks = tiles / WAVES;            // 512 blocks x 256 threads
  band_attn_wmma_kernel<<<blocks, 32 * WAVES, 0, stream>>>(Q, K, V, G, mask, out);
}